// YoloV3LossDena_64845416235381
// MI455X (gfx1250) — compile-verified
//
#include <hip/hip_runtime.h>
#include <stdint.h>

// ---------------------------------------------------------------------------
// YOLOv3 loss on MI455X (gfx1250).
// Bandwidth-bound reduction: only channels 0..4 of pred matter everywhere
// (BCE(0,0)==0 kills masked channels), so we stream ~3.4MB instead of 58MB.
// CDNA5-specific path used: global_load_async_to_lds_b128 (ASYNCcnt) to stage
// the per-batch target table into LDS, wave32-sized blocks, LDS reductions.
// No WMMA: the workload contains no dense matmul content.
// ---------------------------------------------------------------------------

#define BATCH   16
#define NTGT    50
#define NCLS    80
#define NCH     85
#define IGNORE_TH 0.7f

#define PB_BLK  256
#define PB_X    32              // ceil(3*52*52 / 256)
#define NPART   (PB_X * BATCH * 3)

struct __align__(16) Rec {
    float lx, ly, lw, lh;       // label box (cx,cy,w,h) in grid units
    float sval, fx, fy, lwx;    // sqrt(2-w*h), frac xy, log(w/anchor)
    float lwy;
    int   flat;                 // assigned ? (a*g + jj)*g + ii : -1
    int   cls;
    int   valid;
};
static_assert(sizeof(Rec) == 48, "Rec must be 48 bytes (3 x b128 chunks)");

__device__ __forceinline__ float slog_(float x) {
    return logf(fmaxf(x, 1e-12f));
}
__device__ __forceinline__ float bce_(float p, float t) {
    return -(t * slog_(p) + (1.0f - t) * slog_(1.0f - p));
}

// ---------------------------------------------------------------------------
__global__ void init_k(int* __restrict__ has) {
    int i = threadIdx.x;
    if (i < 3 * BATCH) has[i] = 0;
}

// ---------------------------------------------------------------------------
// Phase A: one thread per (layer, batch, target). 2400 threads total.
__global__ void phaseA_k(const float* __restrict__ targets,
                         const float* __restrict__ anchors,      // (3,3,2)
                         const float* __restrict__ ref_anchors,  // (3,9,4)
                         Rec* __restrict__ recs,
                         int* __restrict__ has)
{
    int tid = blockIdx.x * blockDim.x + threadIdx.x;
    const int total = 3 * BATCH * NTGT;
    if (tid >= total) return;
    int n = tid / (BATCH * NTGT);
    int r = tid - n * (BATCH * NTGT);
    int b = r / NTGT;
    // (t = r % NTGT is implicit in the flat record index)

    const int gtab[3] = {13, 26, 52};
    const int   gi = gtab[n];
    const float g  = (float)gi;

    const float* tp = targets + (size_t)r * 5 + (size_t)n * 0; // recs are [n][b][t]
    tp = targets + (size_t)(b * NTGT + (r - b * NTGT)) * 5;
    float c0 = tp[0], cx = tp[1], cy = tp[2], w = tp[3], h = tp[4];
    int valid = (c0 + cx + cy + w + h) > 0.0f;

    float lx = cx * g, ly = cy * g, lw = w * g, lh = h * g;
    float areaA = lw * lh;

    // argmax IoU of [0,0,lw,lh] vs the 9 reference anchors (first max wins)
    int best = 0; float bestv = -1e30f;
    for (int k = 0; k < 9; ++k) {
        float rw = ref_anchors[n * 36 + k * 4 + 2];
        float rh = ref_anchors[n * 36 + k * 4 + 3];
        float bw = fminf(lw, rw), bh = fminf(lh, rh);
        float ai = (bw > 0.0f && bh > 0.0f) ? bw * bh : 0.0f;
        float iou = ai / (areaA + rw * rh - ai);
        if (iou > bestv) { bestv = iou; best = k; }
    }
    int inm    = (best / 3) == (2 - n);   // layer n owns anchor group 2-n
    int best_a = best % 3;
    int assigned = valid && inm;

    int ii = (int)lx, jj = (int)ly;
    int flat = assigned ? (best_a * gi + jj) * gi + ii : -1;

    float aw = anchors[n * 6 + best_a * 2 + 0];
    float ah = anchors[n * 6 + best_a * 2 + 1];

    Rec rec;
    rec.lx = lx; rec.ly = ly; rec.lw = lw; rec.lh = lh;
    rec.sval = sqrtf(2.0f - w * h);
    rec.fx = lx - floorf(lx);
    rec.fy = ly - floorf(ly);
    rec.lwx = logf(lw / aw + 1e-16f);
    rec.lwy = logf(lh / ah + 1e-16f);
    rec.flat = flat;
    rec.cls  = (int)c0;
    rec.valid = valid;
    recs[tid] = rec;

    if (assigned) atomicOr(&has[n * BATCH + b], 1);
}

// ---------------------------------------------------------------------------
// Phase B: grid (PB_X, BATCH, 3); 256 threads (8 wave32s) per block.
// Each block = one (layer, batch) slab of cells; target table async-staged
// into LDS once per block via the CDNA5 ASYNCcnt path.
__global__ void __launch_bounds__(PB_BLK) phaseB_k(
    const float* __restrict__ pred0, const float* __restrict__ pred1,
    const float* __restrict__ pred2, const float* __restrict__ anchors,
    const Rec* __restrict__ recs, const int* __restrict__ has,
    float* __restrict__ part)
{
    __shared__ __align__(16) Rec srecs[NTGT];
    __shared__ float sred[PB_BLK];

    const int n = blockIdx.z;
    const int b = blockIdx.y;
    const int gtab[3] = {13, 26, 52};
    const int g = gtab[n];
    const int cells = 3 * g * g;
    const int pidx = (n * BATCH + b) * PB_X + blockIdx.x;
    const int tid = threadIdx.x;

    if ((int)blockIdx.x * PB_BLK >= cells) {   // uniform whole-block early-out
        if (tid == 0) part[pidx] = 0.0f;
        return;
    }
    const float* pred = (n == 0) ? pred0 : (n == 1) ? pred1 : pred2;

    // ---- async-stage 50 records (2400B = 150 x b128) into LDS -------------
    {
        const int nchunk = (NTGT * (int)sizeof(Rec)) / 16;   // 150
        if (tid < nchunk) {
            uint32_t lds = (uint32_t)(uintptr_t)(&srecs[0]) + (uint32_t)tid * 16u;
            uint64_t ga  = (uint64_t)(uintptr_t)(recs + (size_t)(n * BATCH + b) * NTGT)
                         + (uint64_t)tid * 16u;
            asm volatile("global_load_async_to_lds_b128 %0, %1, off"
                         :: "v"(lds), "v"(ga) : "memory");
        }
        asm volatile("s_wait_asynccnt 0" ::: "memory");
        __syncthreads();
    }

    float loss = 0.0f;
    const int cell = (int)blockIdx.x * PB_BLK + tid;
    if (cell < cells) {
        const int a   = cell / (g * g);
        const int rem = cell - a * g * g;
        const int y   = rem / g;
        const int x   = rem - y * g;
        const float* pp = pred + (size_t)(((b * 3 + a) * g + y) * g + x) * NCH;
        float p0 = pp[0], p1 = pp[1], p2 = pp[2], p3 = pp[3], p4 = pp[4];

        float aw = anchors[n * 6 + a * 2 + 0];
        float ah = anchors[n * 6 + a * 2 + 1];
        float px = p0 + (float)x, py = p1 + (float)y;
        float pw = expf(p2) * aw, ph = expf(p3) * ah;
        float pax0 = px - 0.5f * pw, pax1 = px + 0.5f * pw;
        float pay0 = py - 0.5f * ph, pay1 = py + 0.5f * ph;
        float areaP = pw * ph;

        float best_iou = -1.0f;
        int atgt = -1;                       // last target mapping here wins
        const int myflat = cell;             // == (a*g + y)*g + x
        for (int t = 0; t < NTGT; ++t) {
            Rec rc = srecs[t];
            if (rc.flat == myflat) atgt = t;
            if (rc.valid) {
                float bx0 = rc.lx - 0.5f * rc.lw, bx1 = rc.lx + 0.5f * rc.lw;
                float by0 = rc.ly - 0.5f * rc.lh, by1 = rc.ly + 0.5f * rc.lh;
                float tlx = fmaxf(pax0, bx0), brx = fminf(pax1, bx1);
                float tly = fmaxf(pay0, by0), bry = fminf(pay1, by1);
                float iw = brx - tlx, ih = bry - tly;
                float ai = (iw > 0.0f && ih > 0.0f) ? iw * ih : 0.0f;
                float iou = ai / (areaP + rc.lw * rc.lh - ai);
                best_iou = fmaxf(best_iou, iou);
            }
        }

        if (atgt >= 0) {
            Rec rc = srecs[atgt];
            float s2 = rc.sval * rc.sval;
            loss += s2 * (bce_(p0, rc.fx) + bce_(p1, rc.fy));          // xy
            float dw = p2 - rc.lwx, dh = p3 - rc.lwy;
            loss += 0.5f * s2 * (dw * dw + dh * dh);                   // wh
            loss += -slog_(p4);                                        // obj(t=1)
            for (int c = 0; c < NCLS; ++c) {                           // cls
                float pc = pp[5 + c];
                loss += (c == rc.cls) ? -slog_(pc) : -slog_(1.0f - pc);
            }
        } else {
            int hb = has[n * BATCH + b];
            float obj = hb ? ((best_iou <= IGNORE_TH) ? 1.0f : 0.0f) : 1.0f;
            if (obj != 0.0f) loss += -slog_(1.0f - p4);                // obj(t=0)
        }
    }

    // block reduction (LDS tree), one deterministic partial per block
    sred[tid] = loss;
    __syncthreads();
    for (int st = PB_BLK / 2; st > 0; st >>= 1) {
        if (tid < st) sred[tid] += sred[tid + st];
        __syncthreads();
    }
    if (tid == 0) part[pidx] = sred[0];
}

// ---------------------------------------------------------------------------
// Deterministic final reduction of the NPART block partials.
__global__ void reduce_k(const float* __restrict__ part, float* __restrict__ out)
{
    __shared__ float s[256];
    float v = 0.0f;
    for (int i = threadIdx.x; i < NPART; i += 256) v += part[i];
    s[threadIdx.x] = v;
    __syncthreads();
    for (int st = 128; st > 0; st >>= 1) {
        if (threadIdx.x < st) s[threadIdx.x] += s[threadIdx.x + st];
        __syncthreads();
    }
    if (threadIdx.x == 0) out[0] = s[0];
}

// ---------------------------------------------------------------------------
extern "C" void kernel_launch(void* const* d_in, const int* in_sizes, int n_in,
                              void* d_out, int out_size, void* d_ws, size_t ws_size,
                              hipStream_t stream)
{
    const float* pred0       = (const float*)d_in[0];
    const float* pred1       = (const float*)d_in[1];
    const float* pred2       = (const float*)d_in[2];
    const float* targets     = (const float*)d_in[3];
    const float* anchors     = (const float*)d_in[4];
    const float* ref_anchors = (const float*)d_in[5];
    float* out = (float*)d_out;

    // workspace layout: [recs: 3*16*50*48B][has: 48 ints][partials: NPART floats]
    Rec*   recs = (Rec*)d_ws;
    char*  base = (char*)d_ws + (size_t)3 * BATCH * NTGT * sizeof(Rec);
    int*   has  = (int*)base;
    float* part = (float*)(base + 3 * BATCH * sizeof(int));

    init_k<<<1, 64, 0, stream>>>(has);
    phaseA_k<<<(3 * BATCH * NTGT + 255) / 256, 256, 0, stream>>>(
        targets, anchors, ref_anchors, recs, has);
    dim3 gridB(PB_X, BATCH, 3);
    phaseB_k<<<gridB, PB_BLK, 0, stream>>>(
        pred0, pred1, pred2, anchors, recs, has, part);
    reduce_k<<<1, 256, 0, stream>>>(part, out);
}